// Fusionloss_57741540327685
// MI455X (gfx1250) — compile-verified
//
#include <hip/hip_runtime.h>
#include <math.h>

#define B_   4
#define H_   480
#define W_   640
#define HW   (H_ * W_)
#define NPIX (B_ * HW)          // 1,228,800
#define NBLK 4800               // NPIX/256 == number of 16x16 tiles (4*30*40)

typedef __attribute__((ext_vector_type(2))) float v2f;
typedef __attribute__((ext_vector_type(8))) float v8f;

__device__ __forceinline__ int refl(int i, int n) {
    if (i < 0) i = -i;                 // jnp.pad mode='reflect'
    if (i >= n) i = 2 * n - 2 - i;
    return i;
}

// ---------------------------------------------------------------------------
// Kernel 1: 11x11 bilateral filter of image_ir (reflect pad).
// Fuse spatial+color weights into a single expf: w = exp(-(dx^2+dy^2)/128 - d^2*200)
// ---------------------------------------------------------------------------
__global__ void k_bilateral(const float* __restrict__ ir, float* __restrict__ bilat) {
    int p = blockIdx.x * 256 + threadIdx.x;
    int b = p / HW, rem = p % HW, h = rem / W_, w = rem % W_;
    const float* base = ir + (size_t)b * HW;
    float c = base[h * W_ + w];
    float wsum = 0.f, vsum = 0.f;
    for (int dy = -5; dy <= 5; ++dy) {
        int rr = refl(h + dy, H_) * W_;
        for (int dx = -5; dx <= 5; ++dx) {
            int cc = refl(w + dx, W_);
            float v = base[rr + cc];
            float d = v - c;
            float e = expf(-(float)(dx * dx + dy * dy) * (1.f / 128.f) - d * d * 200.f);
            wsum += e;
            vsum += e * v;
        }
    }
    bilat[p] = vsum / wsum;
}

// ---------------------------------------------------------------------------
// Kernel 2: pointwise losses + Sobel gradient loss, block-reduced partials.
// ---------------------------------------------------------------------------
__device__ __forceinline__ float sobel_abs(const float v[9]) {
    // cross-correlation with _SOBEL_X / _SOBEL_Y (lax.conv does not flip)
    float gx = -v[0] + v[2] - 2.f * v[3] + 2.f * v[5] - v[6] + v[8];
    float gy =  v[0] + 2.f * v[1] + v[2] - v[6] - 2.f * v[7] - v[8];
    return fabsf(gx) + fabsf(gy);
}

__global__ void k_point(const float* __restrict__ vis, const float* __restrict__ ir,
                        const float* __restrict__ gen, const float* __restrict__ bilat,
                        float* __restrict__ p_in, float* __restrict__ p_grad,
                        float* __restrict__ p_tra) {
    int p = blockIdx.x * 256 + threadIdx.x;
    int b = p / HW, rem = p % HW, h = rem / W_, w = rem % W_;
    const float* ybase = vis + (size_t)b * 3 * HW;   // channel 0 = image_y
    const float* ibase = ir    + (size_t)b * HW;
    const float* gbase = gen   + (size_t)b * HW;
    const float* fbase = bilat + (size_t)b * HW;

    float yv = ybase[rem], iv = ibase[rem], gv = gbase[rem];
    float t_in  = fabsf(gv - fmaxf(yv, iv));
    float t_tra = fabsf(gv - 0.5f * (iv + yv));

    float vy[9], vg[9], vf[9];
    int q = 0;
    for (int di = 0; di < 3; ++di) {
        int rr = refl(h + di - 1, H_) * W_;
        for (int dj = 0; dj < 3; ++dj) {
            int o = rr + refl(w + dj - 1, W_);
            vy[q] = ybase[o]; vg[q] = gbase[o]; vf[q] = fbase[o]; ++q;
        }
    }
    float t_grad = fabsf(sobel_abs(vg) - fmaxf(sobel_abs(vy), sobel_abs(vf)));

    __shared__ float r0[256], r1[256], r2[256];
    int t = threadIdx.x;
    r0[t] = t_in; r1[t] = t_grad; r2[t] = t_tra;
    __syncthreads();
    for (int s = 128; s > 0; s >>= 1) {
        if (t < s) { r0[t] += r0[t + s]; r1[t] += r1[t + s]; r2[t] += r2[t + s]; }
        __syncthreads();
    }
    if (t == 0) { p_in[blockIdx.x] = r0[0]; p_grad[blockIdx.x] = r1[0]; p_tra[blockIdx.x] = r2[0]; }
}

// ---------------------------------------------------------------------------
// Kernel 3: SSIM. One wave (32 lanes) per 16x16 output tile.
//   Phase 1: load 26x26 halo (zero pad) of y / ir / gen into LDS.
//   Phase 2: horizontal 11-tap Gaussian over 8 product channels -> LDS (28x16, K-padded).
//   Phase 3: vertical 11-tap pass as banded matmul: 7 chained V_WMMA_F32_16X16X4_F32
//            per channel: D(16x16) = G(16x28) * X(28x16). A operands are hoisted and
//            built branchlessly so EXEC stays all-ones through the WMMA sequence.
//   Phase 4: per-pixel SSIM maps, wave reduce, write tile partial sums.
// ---------------------------------------------------------------------------
__global__ void __launch_bounds__(32)
k_ssim(const float* __restrict__ vis, const float* __restrict__ ir,
       const float* __restrict__ gen, float* __restrict__ p_s1, float* __restrict__ p_s2) {
    __shared__ float sy[26 * 26], si[26 * 26], sg[26 * 26];
    __shared__ float hc[8][28 * 16];
    __shared__ float gsh[11];

    int blk = blockIdx.x;
    int tX = blk % 40, tY = (blk / 40) % 30, b = blk / 1200;
    int lane = threadIdx.x;
    int R0 = tY * 16 - 5, C0 = tX * 16 - 5;

    const float* ybase = vis + (size_t)b * 3 * HW;
    const float* ibase = ir  + (size_t)b * HW;
    const float* gbase = gen + (size_t)b * HW;

    // normalized 11-tap Gaussian, sigma = 1.5
    float g[11];
    float gs = 0.f;
#pragma unroll
    for (int j = 0; j < 11; ++j) { float d = (float)(j - 5); g[j] = expf(-d * d / 4.5f); gs += g[j]; }
    float ginv = 1.f / gs;
#pragma unroll
    for (int j = 0; j < 11; ++j) g[j] *= ginv;
    if (lane == 0) {
#pragma unroll
        for (int j = 0; j < 11; ++j) gsh[j] = g[j];
    }

    // Phase 1: halo load with ZERO padding (lax explicit padding)
    for (int e = lane; e < 26 * 26; e += 32) {
        int rr = e / 26, cc = e % 26;
        int gr = R0 + rr, gc = C0 + cc;
        float yv = 0.f, iv = 0.f, gv = 0.f;
        if (gr >= 0 && gr < H_ && gc >= 0 && gc < W_) {
            int o = gr * W_ + gc;
            yv = ybase[o]; iv = ibase[o]; gv = gbase[o];
        }
        sy[e] = yv; si[e] = iv; sg[e] = gv;
    }
    __syncthreads();

    // Phase 2: horizontal conv over 8 channels, rows 26..27 zero-filled (K pad)
    for (int pos = lane; pos < 28 * 16; pos += 32) {
        int rr = pos / 16, cc = pos % 16;
        float s0 = 0, s1 = 0, s2 = 0, s3 = 0, s4 = 0, s5 = 0, s6 = 0, s7 = 0;
        if (rr < 26) {
#pragma unroll
            for (int j = 0; j < 11; ++j) {
                float wj = g[j];
                int o = rr * 26 + cc + j;
                float yv = sy[o], iv = si[o], gv = sg[o];
                s0 += wj * yv;      s1 += wj * iv;      s2 += wj * gv;
                s3 += wj * yv * yv; s4 += wj * iv * iv; s5 += wj * gv * gv;
                s6 += wj * yv * gv; s7 += wj * iv * gv;
            }
        }
        hc[0][pos] = s0; hc[1][pos] = s1; hc[2][pos] = s2; hc[3][pos] = s3;
        hc[4][pos] = s4; hc[5][pos] = s5; hc[6][pos] = s6; hc[7][pos] = s7;
    }
    __syncthreads();

    // Phase 3: vertical conv via f32 WMMA. ISA layouts (§7.12.2):
    //   A 16x4: M = lane&15; vgpr0->K = 4t+k0, vgpr1->K = 4t+k0+1, k0 = 2*(lane>=16)
    //   B 4x16: N = lane&15; vgpr r holds rows K = 4t+r (lanes 0-15) / 4t+r+2 (lanes 16-31)
    //   D 16x16: vgpr r -> M = r + 8*(lane>=16), N = lane&15
    int m_ = lane & 15;
    int k0 = (lane >> 4) << 1;

    // Hoisted, branchless A operands: unconditional clamped LDS read + mask select.
    v2f av[7];
#pragma unroll
    for (int t = 0; t < 7; ++t) {
        int ka = 4 * t + k0;
        int d0 = ka - m_, d1 = d0 + 1;            // banded G[m][k] = g[k-m]
        int i0 = min(max(d0, 0), 10);
        int i1 = min(max(d1, 0), 10);
        float f0 = gsh[i0];
        float f1 = gsh[i1];
        av[t].x = (d0 >= 0 && d0 <= 10) ? f0 : 0.f;
        av[t].y = (d1 >= 0 && d1 <= 10) ? f1 : 0.f;
    }

    v8f acc[8] = {};
#pragma unroll
    for (int t = 0; t < 7; ++t) {
        int ka = 4 * t + k0;
#pragma unroll
        for (int ch = 0; ch < 8; ++ch) {
            v2f bb;
            bb.x = hc[ch][ka * 16 + m_];
            bb.y = hc[ch][(ka + 1) * 16 + m_];
            acc[ch] = __builtin_amdgcn_wmma_f32_16x16x4_f32(
                false, av[t], false, bb, (short)0, acc[ch], false, false);
        }
    }

    // Phase 4: SSIM maps and reduction (positions irrelevant: we only need the sum)
    const float C1 = 1e-4f, C2 = 9e-4f;
    float ss1 = 0.f, ss2 = 0.f;
#pragma unroll
    for (int r = 0; r < 8; ++r) {
        float muy = acc[0][r], mui = acc[1][r], mug = acc[2][r];
        float eyy = acc[3][r], eii = acc[4][r], egg = acc[5][r];
        float eyg = acc[6][r], eig = acc[7][r];
        float sgg = egg - mug * mug;
        {
            float s11 = eyy - muy * muy;
            float s12 = eyg - muy * mug;
            ss1 += ((2.f * muy * mug + C1) * (2.f * s12 + C2)) /
                   ((muy * muy + mug * mug + C1) * (s11 + sgg + C2));
        }
        {
            float s11 = eii - mui * mui;
            float s12 = eig - mui * mug;
            ss2 += ((2.f * mui * mug + C1) * (2.f * s12 + C2)) /
                   ((mui * mui + mug * mug + C1) * (s11 + sgg + C2));
        }
    }
    for (int off = 16; off > 0; off >>= 1) {
        ss1 += __shfl_down(ss1, off, 32);
        ss2 += __shfl_down(ss2, off, 32);
    }
    if (lane == 0) { p_s1[blk] = ss1; p_s2[blk] = ss2; }
}

// ---------------------------------------------------------------------------
// Kernel 4: deterministic final reduce of the 5 partial arrays + loss combine.
// ---------------------------------------------------------------------------
__global__ void k_final(const float* __restrict__ p_in, const float* __restrict__ p_grad,
                        const float* __restrict__ p_tra, const float* __restrict__ p_s1,
                        const float* __restrict__ p_s2, float* __restrict__ out) {
    __shared__ float red[256];
    int t = threadIdx.x;
    const float* arrs[5] = { p_in, p_grad, p_tra, p_s1, p_s2 };
    float res[5];
    for (int a = 0; a < 5; ++a) {
        float s = 0.f;
        for (int i = t; i < NBLK; i += 256) s += arrs[a][i];
        red[t] = s;
        __syncthreads();
        for (int sft = 128; sft > 0; sft >>= 1) {
            if (t < sft) red[t] += red[t + sft];
            __syncthreads();
        }
        res[a] = red[0];
        __syncthreads();
    }
    if (t == 0) {
        const float invN = 1.f / (float)NPIX;
        float loss_in   = res[0] * invN;
        float loss_grad = res[1] * invN;
        float loss_tra  = res[2] * invN;
        float m1 = res[3] * invN, m2 = res[4] * invN;
        float loss_ssim = (1.f - m1) * 0.5f + (1.f - m2) * 0.5f;
        float total = 10.f * loss_in + 35.f * loss_grad + 1.f * loss_ssim + 10.f * loss_tra;
        out[0] = total; out[1] = loss_in; out[2] = loss_grad; out[3] = loss_ssim; out[4] = loss_tra;
    }
}

// ---------------------------------------------------------------------------
extern "C" void kernel_launch(void* const* d_in, const int* in_sizes, int n_in,
                              void* d_out, int out_size, void* d_ws, size_t ws_size,
                              hipStream_t stream) {
    (void)in_sizes; (void)n_in; (void)out_size; (void)ws_size;
    const float* vis = (const float*)d_in[0];  // (4,3,480,640)
    const float* ir  = (const float*)d_in[1];  // (4,1,480,640)
    const float* gen = (const float*)d_in[2];  // (4,1,480,640)
    float* out = (float*)d_out;                // 5 scalars

    float* wsf    = (float*)d_ws;
    float* bilat  = wsf;                 // NPIX floats
    float* p_in   = wsf + NPIX;          // NBLK floats each
    float* p_grad = p_in + NBLK;
    float* p_tra  = p_grad + NBLK;
    float* p_s1   = p_tra + NBLK;
    float* p_s2   = p_s1 + NBLK;

    k_bilateral<<<NBLK, 256, 0, stream>>>(ir, bilat);
    k_point   <<<NBLK, 256, 0, stream>>>(vis, ir, gen, bilat, p_in, p_grad, p_tra);
    k_ssim    <<<NBLK, 32,  0, stream>>>(vis, ir, gen, p_s1, p_s2);
    k_final   <<<1,    256, 0, stream>>>(p_in, p_grad, p_tra, p_s1, p_s2, out);
}